// SSIM1d_74861279969541
// MI455X (gfx1250) — compile-verified
//
#include <hip/hip_runtime.h>

#define NTOT   16777216
#define WIN    7
#define MOUT   (NTOT - WIN + 1)          // 16777210 outputs
#define THREADS 256
#define OPB    2048                       // outputs per block (kssim)
#define OPT    8                          // outputs per thread
#define TILE_IN (OPB + 16)                // 2064 floats staged per array
#define NB_MAX 1024
#define NB_SSIM ((MOUT + OPB - 1) / OPB)  // 8192

typedef float v2f __attribute__((ext_vector_type(2)));
typedef float v8f __attribute__((ext_vector_type(8)));
typedef int   v4i __attribute__((ext_vector_type(4)));
typedef __attribute__((address_space(1))) v4i as1_v4i;  // global int4
typedef __attribute__((address_space(3))) v4i as3_v4i;  // LDS int4

#if __has_builtin(__builtin_amdgcn_global_load_async_to_lds_b128)
#define HAVE_ASYNC 1
#else
#define HAVE_ASYNC 0
#endif

#if __has_builtin(__builtin_amdgcn_wmma_f32_16x16x4_f32)
#define HAVE_WMMA 1
#else
#define HAVE_WMMA 0
#endif

#if HAVE_ASYNC
__device__ __forceinline__ void async_wait0() {
#if __has_builtin(__builtin_amdgcn_s_wait_asynccnt)
  __builtin_amdgcn_s_wait_asynccnt(0);
#else
  asm volatile("s_wait_asynccnt 0x0" ::: "memory");
#endif
}
__device__ __forceinline__ as1_v4i* to_global_v4i(const void* p) {
  return (as1_v4i*)(unsigned long long)p;
}
__device__ __forceinline__ as3_v4i* to_lds_v4i(const void* p) {
  return (as3_v4i*)(unsigned)(unsigned long long)p;
}
#endif

// ---------------- pass 1: per-block max of y ----------------
__global__ void kmax(const float* __restrict__ y, float* __restrict__ wmax) {
  int tid = threadIdx.x, bid = blockIdx.x;
  const float4* y4 = (const float4*)y;
  float m = -__builtin_inff();
  for (int i = bid * THREADS + tid; i < NTOT / 4; i += NB_MAX * THREADS) {
    float4 v = y4[i];
    m = fmaxf(m, fmaxf(fmaxf(v.x, v.y), fmaxf(v.z, v.w)));
  }
#pragma unroll
  for (int off = 16; off; off >>= 1) m = fmaxf(m, __shfl_xor(m, off, 32));
  __shared__ float s[8];
  if ((tid & 31) == 0) s[tid >> 5] = m;
  __syncthreads();
  if (tid == 0) {
    float mm = s[0];
#pragma unroll
    for (int k = 1; k < 8; ++k) mm = fmaxf(mm, s[k]);
    wmax[bid] = mm;
  }
}

// ---------------- pass 2: final max -> ws[0] ----------------
__global__ void kmax2(const float* __restrict__ wmax, float* __restrict__ wdr) {
  int tid = threadIdx.x;
  float m = -__builtin_inff();
  for (int i = tid; i < NB_MAX; i += THREADS) m = fmaxf(m, wmax[i]);
#pragma unroll
  for (int off = 16; off; off >>= 1) m = fmaxf(m, __shfl_xor(m, off, 32));
  __shared__ float s[8];
  if ((tid & 31) == 0) s[tid >> 5] = m;
  __syncthreads();
  if (tid == 0) {
    float mm = s[0];
#pragma unroll
    for (int k = 1; k < 8; ++k) mm = fmaxf(mm, s[k]);
    wdr[0] = mm;
  }
}

// ---------------- pass 3: SSIM partial sums ----------------
__global__ void kssim(const float* __restrict__ x, const float* __restrict__ y,
                      const float* __restrict__ wdr, float* __restrict__ wsum) {
  __shared__ __align__(16) float lx[TILE_IN];
  __shared__ __align__(16) float ly[TILE_IN];
  __shared__ float sred[THREADS];

  const int tid = threadIdx.x;
  const long long t0 = (long long)blockIdx.x * OPB;

  // stage tile (coalesced, float4 granularity); tail block zero-fills
  const int nvec = TILE_IN / 4;
  for (int v = tid; v < nvec; v += THREADS) {
    long long gi = t0 + (long long)v * 4;
    if (gi + 3 < NTOT) {
#if HAVE_ASYNC
      __builtin_amdgcn_global_load_async_to_lds_b128(
          to_global_v4i(x + gi), to_lds_v4i(&lx[v * 4]), 0, 0);
      __builtin_amdgcn_global_load_async_to_lds_b128(
          to_global_v4i(y + gi), to_lds_v4i(&ly[v * 4]), 0, 0);
#else
      *(float4*)&lx[v * 4] = *(const float4*)(x + gi);
      *(float4*)&ly[v * 4] = *(const float4*)(y + gi);
#endif
    } else {
#pragma unroll
      for (int e = 0; e < 4; ++e) {
        long long g = gi + e;
        lx[v * 4 + e] = (g < NTOT) ? x[g] : 0.0f;
        ly[v * 4 + e] = (g < NTOT) ? y[g] : 0.0f;
      }
    }
  }
#if HAVE_ASYNC
  async_wait0();
#endif
  __syncthreads();

  const float dr   = wdr[0];
  const float C1   = (0.01f * dr) * (0.01f * dr);
  const float C2   = (0.03f * dr) * (0.03f * dr);
  const float inv7 = 1.0f / 7.0f;
  const float covn = 7.0f / 6.0f;

  const int ofs = tid * OPT;          // first output of this thread within tile
  const long long obase = t0 + ofs;   // global output index

  float sx = 0.f, sy = 0.f, sxx = 0.f, syy = 0.f, sxy = 0.f;
#pragma unroll
  for (int k = 0; k < WIN; ++k) {
    float a = lx[ofs + k], b = ly[ofs + k];
    sx += a; sy += b; sxx += a * a; syy += b * b; sxy += a * b;
  }

  float local = 0.f;
#pragma unroll
  for (int r = 0; r < OPT; ++r) {
    if (obase + r < MOUT) {
      float ux = sx * inv7, uy = sy * inv7;
      float uxx = sxx * inv7, uyy = syy * inv7, uxy = sxy * inv7;
      float vx  = covn * (uxx - ux * ux);
      float vy  = covn * (uyy - uy * uy);
      float vxy = covn * (uxy - ux * uy);
      float A1 = 2.0f * ux * uy + C1;
      float A2 = 2.0f * vxy + C2;
      float B1 = ux * ux + uy * uy + C1;
      float B2 = vx + vy + C2;
      local += (A1 * A2) / (B1 * B2);
    }
    if (r < OPT - 1) {  // slide window by one
      float ao = lx[ofs + r],       bo = ly[ofs + r];
      float an = lx[ofs + r + WIN], bn = ly[ofs + r + WIN];
      sx  += an - ao;            sy  += bn - bo;
      sxx += an * an - ao * ao;  syy += bn * bn - bo * bo;
      sxy += an * bn - ao * bo;
    }
  }

  sred[tid] = local;
  __syncthreads();

  if (tid < 32) {  // wave 0, EXEC all ones
    float t;
#if HAVE_WMMA
    // Sum 256 values on the matrix unit: D = A x Ones(+C) => sum(D) = 16*sum(A)
    v8f acc = {0.f, 0.f, 0.f, 0.f, 0.f, 0.f, 0.f, 0.f};
    v2f ones; ones[0] = 1.0f; ones[1] = 1.0f;
#pragma unroll
    for (int c = 0; c < THREADS / 64; ++c) {
      v2f a; a[0] = sred[c * 64 + tid]; a[1] = sred[c * 64 + 32 + tid];
      acc = __builtin_amdgcn_wmma_f32_16x16x4_f32(false, a, false, ones,
                                                  (short)0, acc, false, false);
    }
    t = ((acc[0] + acc[1]) + (acc[2] + acc[3])) +
        ((acc[4] + acc[5]) + (acc[6] + acc[7]));
#pragma unroll
    for (int off = 16; off; off >>= 1) t += __shfl_xor(t, off, 32);
    t *= (1.0f / 16.0f);
#else
    t = 0.f;
#pragma unroll
    for (int c = 0; c < 8; ++c) t += sred[tid * 8 + c];
#pragma unroll
    for (int off = 16; off; off >>= 1) t += __shfl_xor(t, off, 32);
#endif
    if (tid == 0) wsum[blockIdx.x] = t;
  }
}

// ---------------- pass 4: final mean ----------------
__global__ void kfinal(const float* __restrict__ wsum, float* __restrict__ out) {
  int tid = threadIdx.x;
  double s = 0.0;
  for (int i = tid; i < NB_SSIM; i += THREADS) s += (double)wsum[i];
#pragma unroll
  for (int off = 16; off; off >>= 1) s += __shfl_xor(s, off, 32);
  __shared__ double sd[8];
  if ((tid & 31) == 0) sd[tid >> 5] = s;
  __syncthreads();
  if (tid == 0) {
    double t = 0.0;
#pragma unroll
    for (int k = 0; k < 8; ++k) t += sd[k];
    out[0] = (float)(t / (double)MOUT);
  }
}

extern "C" void kernel_launch(void* const* d_in, const int* in_sizes, int n_in,
                              void* d_out, int out_size, void* d_ws, size_t ws_size,
                              hipStream_t stream) {
  const float* x = (const float*)d_in[0];
  const float* y = (const float*)d_in[1];
  float* out = (float*)d_out;

  float* w    = (float*)d_ws;
  float* wdr  = w;                  // [0]        data_range
  float* wmax = w + 16;             // [16..)     1024 block maxima
  float* wsum = w + 16 + NB_MAX;    // [1040..)   8192 block partial sums

  kmax  <<<NB_MAX,  THREADS, 0, stream>>>(y, wmax);
  kmax2 <<<1,       THREADS, 0, stream>>>(wmax, wdr);
  kssim <<<NB_SSIM, THREADS, 0, stream>>>(x, y, wdr, wsum);
  kfinal<<<1,       THREADS, 0, stream>>>(wsum, out);
}